// Embedding_Table_Group_10342281249257
// MI455X (gfx1250) — compile-verified
//
#include <hip/hip_runtime.h>

// Embedding-bag sum (DLRM-style) on MI455X / gfx1250.
// Segment-sum expressed as WMMA: Out[bag,m] = sum_p S[bag,p]*Rows[p,m],
// computed with V_WMMA_F32_16X16X4_F32 (A = one-hot bag selector, B = gathered rows).
//
// T=26 tables, N=100000 rows, M=128 dim, B=4096 bags, L=32 indices/bag.
// Memory-bound: ~1.74 GB gathered reads -> ~77 us floor at 23.3 TB/s HBM.
//
// Addressing: per-table base pointer is wave-uniform; all divergent math is a
// 32-bit byte offset (row*512 + col*4, max 51.2 MB) so loads lower to the
// SADDR form (global_load_b32 v, voff, s[base]) with one v_add_nc_u32 each.

typedef __attribute__((ext_vector_type(2))) float v2f;
typedef __attribute__((ext_vector_type(8))) float v8f;

#define T_TABLES 26
#define N_ROWS   100000
#define M_DIM    128
#define B_BAGS   4096
#define L_POOL   32
#define BAGS_PER_BLOCK 16   // one 16x16 WMMA accumulator covers 16 bags x 16 cols

__global__ __launch_bounds__(256) void embag_wmma_kernel(
    const int*   __restrict__ lS_i,     // [T, B*L]
    const float* __restrict__ weights,  // [T, N, M]
    float*       __restrict__ out)      // [T, B, M]
{
    __shared__ unsigned int sOff[BAGS_PER_BLOCK * L_POOL];  // row BYTE offsets, 2 KB

    const int tid   = threadIdx.x;
    const int wave  = tid >> 5;
    const int lane  = tid & 31;
    const int laneN = lane & 15;       // N coordinate in the 16x16 tile
    const int half  = lane >> 4;       // selects K pair (2*half, 2*half+1) / upper M half

    const int groupsPerTable = B_BAGS / BAGS_PER_BLOCK;      // 256
    const int t  = blockIdx.x / groupsPerTable;
    const int g0 = (blockIdx.x % groupsPerTable) * BAGS_PER_BLOCK;

    // Stage this block's 512 bag indices into LDS as row byte offsets
    // (index * M*4 = index << 9); each thread converts 2.
    const int* gi = lS_i + (size_t)t * (B_BAGS * L_POOL) + (size_t)g0 * L_POOL;
    sOff[tid]       = ((unsigned int)gi[tid])       << 9;
    sOff[tid + 256] = ((unsigned int)gi[tid + 256]) << 9;
    __syncthreads();

    // Wave-uniform table base; divergent part stays in a 32-bit byte offset.
    const char* wt = (const char*)(weights + (size_t)t * N_ROWS * M_DIM);
    const unsigned int col4 = (unsigned int)(wave * 16 + laneN) * 4u;

    // Two accumulators to relax the WMMA RAW chain; summed at the end.
    v8f acc0 = {};
    v8f acc1 = {};

    for (int g = 0; g < BAGS_PER_BLOCK; ++g) {
        // A (16x4 selector): ones in matrix row M == g, zero elsewhere.
        // A layout: lanes 0-15 M=lane, lanes 16-31 M=lane-16;
        // VGPR0 holds K=0/K=2, VGPR1 holds K=1/K=3 -> both equal here.
        const float aval = (laneN == g) ? 1.0f : 0.0f;
        v2f A; A.x = aval; A.y = aval;

        const int pbase = g * L_POOL;
        #pragma unroll
        for (int cc = 0; cc < L_POOL / 4; ++cc) {
            // B (4x16): VGPR0 = rows K0 (lanes 0-15) / K2 (lanes 16-31),
            //           VGPR1 = rows K1 / K3, N = laneN.
            const int p0 = pbase + cc * 4 + half * 2;
            const unsigned int o0 = sOff[p0]     + col4;   // K = half*2
            const unsigned int o1 = sOff[p0 + 1] + col4;   // K = half*2 + 1
            v2f Bv;
            Bv.x = *(const float*)(wt + o0);
            Bv.y = *(const float*)(wt + o1);

            if (cc & 1)
                acc1 = __builtin_amdgcn_wmma_f32_16x16x4_f32(
                          false, A, false, Bv, (short)0, acc1, false, false);
            else
                acc0 = __builtin_amdgcn_wmma_f32_16x16x4_f32(
                          false, A, false, Bv, (short)0, acc0, false, false);
        }
    }

    // D layout (16x16 f32): VGPR r -> M = r + 8*half, N = laneN.
    float* outT = out + ((size_t)t * B_BAGS + g0) * M_DIM;
    const int col = wave * 16 + laneN;
    #pragma unroll
    for (int r = 0; r < 8; ++r) {
        const int bag = r + half * 8;
        outT[(size_t)bag * M_DIM + col] = acc0[r] + acc1[r];
    }
}

extern "C" void kernel_launch(void* const* d_in, const int* in_sizes, int n_in,
                              void* d_out, int out_size, void* d_ws, size_t ws_size,
                              hipStream_t stream) {
    // setup_inputs() order: lS_o (unused: offsets are uniform b*L), lS_i, weights
    const int*   lS_i    = (const int*)d_in[1];
    const float* weights = (const float*)d_in[2];
    float*       out     = (float*)d_out;

    const int nBlocks = T_TABLES * (B_BAGS / BAGS_PER_BLOCK);   // 26*256 = 6656
    embag_wmma_kernel<<<dim3(nBlocks), dim3(256), 0, stream>>>(lS_i, weights, out);
}